// CBMIL_34488587387153
// MI455X (gfx1250) — compile-verified
//
#include <hip/hip_runtime.h>
#include <stdint.h>

#define CC 8
#define NN 16384
#define DD 1024
#define DQ 128
#define KK 128

typedef __attribute__((ext_vector_type(2))) float v2f;
typedef __attribute__((ext_vector_type(8))) float v8f;

__device__ __forceinline__ unsigned f2ord(float f) {
    unsigned u = __float_as_uint(f);
    return (u & 0x80000000u) ? ~u : (u | 0x80000000u);
}
__device__ __forceinline__ float ord2f(unsigned u) {
    unsigned b = (u & 0x80000000u) ? (u & 0x7FFFFFFFu) : ~u;
    return __uint_as_float(b);
}

// ---------------------------------------------------------------- Kernel A
// Per cluster: qk = key@Wq + bq ; b0 = bq.qk ; wvec = Wq@qk ; init umax.
__global__ __launch_bounds__(128) void prep_kernel(
    const float* __restrict__ keyf, const float* __restrict__ Wq,
    const float* __restrict__ bq, float* __restrict__ qk,
    float* __restrict__ wvec, float* __restrict__ b0, unsigned* __restrict__ umax)
{
    const int c = blockIdx.x, j = threadIdx.x;
    __shared__ float qks[DQ];
    __shared__ float red[DQ];
    const float* kf = keyf + (size_t)c * DD;
    float acc = bq[j];
    for (int d = 0; d < DD; ++d) acc += kf[d] * Wq[(size_t)d * DQ + j];
    qks[j] = acc;
    qk[c * DQ + j] = acc;
    red[j] = acc * bq[j];
    __syncthreads();
    for (int s = 64; s > 0; s >>= 1) { if (j < s) red[j] += red[j + s]; __syncthreads(); }
    if (j == 0) { b0[c] = red[0]; umax[c] = 0u; }
    __syncthreads();
    for (int t = 0; t < 8; ++t) {
        int d = t * DQ + j;
        float a = 0.f;
        #pragma unroll 4
        for (int jj = 0; jj < DQ; ++jj) a += Wq[(size_t)d * DQ + jj] * qks[jj];
        wvec[(size_t)c * DD + d] = a;
    }
}

// ---------------------------------------------------------------- Kernel B
// s[c][n] = feats[c][n].wvec[c] + b0[c]; store ordered keys; atomicMax.
__global__ __launch_bounds__(256) void score_kernel(
    const float* __restrict__ feats, const float* __restrict__ wvec,
    const float* __restrict__ b0v, float* __restrict__ scores,
    unsigned* __restrict__ ukeys, unsigned* __restrict__ umax)
{
    __shared__ float wl[DD];
    const int rb = blockIdx.x * 8;
    const int c = rb / NN;                    // NN % 8 == 0 -> uniform per block
    ((float4*)wl)[threadIdx.x] = ((const float4*)(wvec + (size_t)c * DD))[threadIdx.x];
    __syncthreads();
    const int wave = threadIdx.x >> 5, lane = threadIdx.x & 31;
    const int n = (rb % NN) + wave;
    const float4* row = (const float4*)(feats + ((size_t)c * NN + n) * DD);
    const float4* wr = (const float4*)wl;
    float dotv = 0.f;
    #pragma unroll
    for (int j = 0; j < 8; ++j) {
        float4 f = row[lane + j * 32];
        float4 w = wr[lane + j * 32];
        dotv += f.x * w.x + f.y * w.y + f.z * w.z + f.w * w.w;
    }
    #pragma unroll
    for (int d = 16; d > 0; d >>= 1) dotv += __shfl_down(dotv, d, 32);
    if (lane == 0) {
        float s = dotv + b0v[c];
        size_t o = (size_t)c * NN + n;
        scores[o] = s;
        unsigned u = f2ord(s);
        ukeys[o] = u;
        atomicMax(&umax[c], u);
    }
}

// ---------------------------------------------------------------- Kernel C
// Z[c] = sum_n exp((s-m)/sqrt(DQ))
__global__ __launch_bounds__(1024) void sumexp_kernel(
    const float* __restrict__ scores, const unsigned* __restrict__ umax,
    float* __restrict__ Zsum)
{
    const int c = blockIdx.x, t = threadIdx.x;
    __shared__ float red[1024];
    const float m = ord2f(umax[c]);
    const float invsqDQ = 0.08838834764831845f;   // 1/sqrt(128)
    const float* sc = scores + (size_t)c * NN;
    float acc = 0.f;
    #pragma unroll 4
    for (int i = 0; i < 16; ++i) acc += expf((sc[t + i * 1024] - m) * invsqDQ);
    red[t] = acc;
    __syncthreads();
    for (int s = 512; s > 0; s >>= 1) { if (t < s) red[t] += red[t + s]; __syncthreads(); }
    if (t == 0) Zsum[c] = red[0];
}

// ---------------------------------------------------------------- Kernel D
// Radix-select K-th largest, gather (ties index-ascending), bitonic sort
// descending, softmax-of-topvals weights A_.
__global__ __launch_bounds__(1024) void topk_kernel(
    const float* __restrict__ scores, const unsigned* __restrict__ ukeys,
    const unsigned* __restrict__ umax, const float* __restrict__ Zsum,
    int* __restrict__ topidx, float* __restrict__ Aw)
{
    const int c = blockIdx.x, t = threadIdx.x;
    const int lane = t & 31, wv = t >> 5;
    __shared__ int redI[32];
    __shared__ int scan[1024];
    __shared__ int idxbuf[KK];
    __shared__ unsigned long long sk[KK];
    __shared__ float pv[KK];
    __shared__ float fred[KK];
    __shared__ int s_total, s_ctr;

    const unsigned* uk = ukeys + (size_t)c * NN;
    unsigned uloc[16];
    #pragma unroll
    for (int i = 0; i < 16; ++i) uloc[i] = uk[t * 16 + i];

    // --- radix select K-th largest ordered key ---
    unsigned prefix = 0u;
    for (int bit = 31; bit >= 0; --bit) {
        const unsigned cand = prefix | (1u << bit);
        const unsigned mask = ~((1u << bit) - 1u);
        int cnt = 0;
        #pragma unroll
        for (int i = 0; i < 16; ++i) cnt += ((uloc[i] & mask) >= cand) ? 1 : 0;
        #pragma unroll
        for (int d = 16; d > 0; d >>= 1) cnt += __shfl_down(cnt, d, 32);
        if (lane == 0) redI[wv] = cnt;
        __syncthreads();
        if (wv == 0) {
            int v = redI[lane];
            #pragma unroll
            for (int d = 16; d > 0; d >>= 1) v += __shfl_down(v, d, 32);
            if (lane == 0) s_total = v;
        }
        __syncthreads();
        if (s_total >= KK) prefix = cand;
    }
    const unsigned T = prefix;

    // --- G = #(u > T) ---
    {
        int cnt = 0;
        #pragma unroll
        for (int i = 0; i < 16; ++i) cnt += (uloc[i] > T) ? 1 : 0;
        #pragma unroll
        for (int d = 16; d > 0; d >>= 1) cnt += __shfl_down(cnt, d, 32);
        if (lane == 0) redI[wv] = cnt;
        __syncthreads();
        if (wv == 0) {
            int v = redI[lane];
            #pragma unroll
            for (int d = 16; d > 0; d >>= 1) v += __shfl_down(v, d, 32);
            if (lane == 0) s_total = v;
        }
        __syncthreads();
    }
    const int G = s_total;
    if (t == 0) s_ctr = 0;
    __syncthreads();

    // gather strictly-greater (set is deterministic; order fixed by sort below)
    #pragma unroll
    for (int i = 0; i < 16; ++i)
        if (uloc[i] > T) { int p = atomicAdd(&s_ctr, 1); idxbuf[p] = t * 16 + i; }

    // ties: exclusive scan over per-thread tie counts (chunks are contiguous)
    int tcnt = 0;
    #pragma unroll
    for (int i = 0; i < 16; ++i) tcnt += (uloc[i] == T) ? 1 : 0;
    scan[t] = tcnt;
    __syncthreads();
    for (int off = 1; off < 1024; off <<= 1) {
        int v = (t >= off) ? scan[t - off] : 0;
        __syncthreads();
        scan[t] += v;
        __syncthreads();
    }
    int run = scan[t] - tcnt;
    const int need = KK - G;
    #pragma unroll
    for (int i = 0; i < 16; ++i) {
        if (uloc[i] == T) {
            if (run < need) idxbuf[G + run] = t * 16 + i;
            ++run;
        }
    }
    __syncthreads();

    // --- bitonic sort: key = (u desc, index asc) ---
    if (t < KK) {
        int n = idxbuf[t];
        sk[t] = ((unsigned long long)uk[n] << 32) | (unsigned)(~(unsigned)n);
    }
    __syncthreads();
    for (int size = 2; size <= KK; size <<= 1) {
        for (int stride = size >> 1; stride > 0; stride >>= 1) {
            if (t < KK) {
                int j = t ^ stride;
                if (j > t) {
                    bool desc = ((t & size) == 0);
                    unsigned long long a = sk[t], b = sk[j];
                    if (desc ? (a < b) : (a > b)) { sk[t] = b; sk[j] = a; }
                }
            }
            __syncthreads();
        }
    }

    // --- top_vals -> second softmax weights ---
    const float invsqDQ = 0.08838834764831845f;  // 1/sqrt(128)
    const float invsqD = 0.03125f;               // 1/sqrt(1024)
    const float m = ord2f(umax[c]);
    const float Z = Zsum[c];
    if (t < KK) {
        int n = (int)(~((unsigned)(sk[t] & 0xFFFFFFFFull)));
        topidx[c * KK + t] = n;
        pv[t] = expf((scores[(size_t)c * NN + n] - m) * invsqDQ) / Z;
    }
    __syncthreads();
    const float maxq = pv[0] * invsqD;           // sorted descending -> pv[0] is max
    float e = 0.f;
    if (t < KK) { e = expf(pv[t] * invsqD - maxq); fred[t] = e; }
    __syncthreads();
    for (int s = 64; s > 0; s >>= 1) { if (t < s) fred[t] += fred[t + s]; __syncthreads(); }
    if (t < KK) Aw[c * KK + t] = e / fred[0];
}

// ---------------------------------------------------------------- Kernel E
// selected_features[c*K+k] = feats[c][topidx[c][k]]
__global__ __launch_bounds__(256) void gather_kernel(
    const float* __restrict__ feats, const int* __restrict__ topidx,
    float* __restrict__ out)
{
    const int row = blockIdx.x;
    const int c = row >> 7, k = row & 127;
    const int n = topidx[c * KK + k];
    const float4* src = (const float4*)(feats + ((size_t)c * NN + n) * DD);
    ((float4*)(out + (size_t)row * DD))[threadIdx.x] = src[threadIdx.x];
}

// ---------------------------------------------------------------- Kernel F
// fusion[c] = A_^T (TF @ Wv + bv) via V_WMMA_F32_16X16X4_F32.
// One wave owns a 16-wide N strip; 8 accumulator tiles cover full M=128.
__global__ __launch_bounds__(256) void fuse_kernel(
    const float* __restrict__ TF, const float* __restrict__ Wv,
    const float* __restrict__ bv, const float* __restrict__ Aw,
    float* __restrict__ fusion)
{
    const int c = blockIdx.y;
    const int wv = threadIdx.x >> 5, lane = threadIdx.x & 31;
    const int n0 = (blockIdx.x * 8 + wv) * 16;
    const int mrow = lane & 15;            // A-frag: lanes 0-15 / 16-31 hold K pairs
    const int koff = (lane >> 4) << 1;     // 0 or 2
    const float* Abase = TF + ((size_t)c * KK + mrow) * DD + koff;
    const float* Bbase = Wv + (size_t)koff * DD + n0 + (lane & 15);

    v8f acc[8];
    #pragma unroll
    for (int mt = 0; mt < 8; ++mt) acc[mt] = (v8f)(0.f);

    for (int k0 = 0; k0 < DD; k0 += 4) {
        v2f b;
        b.x = Bbase[(size_t)k0 * DD];              // Wv[k0+koff  ][n]
        b.y = Bbase[(size_t)k0 * DD + DD];         // Wv[k0+koff+1][n]
        #pragma unroll
        for (int mt = 0; mt < 8; ++mt) {
            v2f a = *(const v2f*)(Abase + (size_t)mt * 16 * DD + k0);
            acc[mt] = __builtin_amdgcn_wmma_f32_16x16x4_f32(
                false, a, false, b, (short)0, acc[mt], false, false);
        }
    }

    // epilogue: partial_n = sum_m A_[m]*(V[m][n]+bv[n]); two lanes per column n
    const float bvn = bv[n0 + (lane & 15)];
    float partial = 0.f;
    #pragma unroll
    for (int mt = 0; mt < 8; ++mt) {
        const int mbase = mt * 16 + ((lane >> 4) << 3);  // hi half-wave -> rows +8
        #pragma unroll
        for (int i = 0; i < 8; ++i)
            partial += Aw[c * KK + mbase + i] * (acc[mt][i] + bvn);
    }
    float other = __shfl_down(partial, 16, 32);
    if (lane < 16) fusion[(size_t)c * DD + n0 + lane] = partial + other;
}

// ---------------------------------------------------------------- launch
extern "C" void kernel_launch(void* const* d_in, const int* in_sizes, int n_in,
                              void* d_out, int out_size, void* d_ws, size_t ws_size,
                              hipStream_t stream) {
    const float* feats = (const float*)d_in[0];   // [C,N,D]
    const float* keyf  = (const float*)d_in[1];   // [C,1,D]
    const float* Wq    = (const float*)d_in[2];   // [D,DQ]
    const float* bq    = (const float*)d_in[3];   // [DQ]
    const float* Wv    = (const float*)d_in[4];   // [D,D]
    const float* bv    = (const float*)d_in[5];   // [D]

    float* selected = (float*)d_out;                          // [C*K, D]
    float* fusion   = (float*)d_out + (size_t)CC * KK * DD;   // [C, D]

    char* w = (char*)d_ws;
    float*    qk     = (float*)(w + 0);          //   4096 B
    float*    wvec   = (float*)(w + 4096);       //  32768 B
    float*    b0     = (float*)(w + 36864);      //     32 B
    unsigned* umax   = (unsigned*)(w + 36896);   //     32 B
    float*    Zsum   = (float*)(w + 36928);      //     32 B
    float*    scores = (float*)(w + 40960);      // 524288 B
    unsigned* ukeys  = (unsigned*)(w + 565248);  // 524288 B
    int*      topidx = (int*)(w + 1089536);      //   4096 B
    float*    Aw     = (float*)(w + 1093632);    //   4096 B  (total ~1.07 MB)

    prep_kernel<<<CC, 128, 0, stream>>>(keyf, Wq, bq, qk, wvec, b0, umax);
    score_kernel<<<(CC * NN) / 8, 256, 0, stream>>>(feats, wvec, b0, scores, ukeys, umax);
    sumexp_kernel<<<CC, 1024, 0, stream>>>(scores, umax, Zsum);
    topk_kernel<<<CC, 1024, 0, stream>>>(scores, ukeys, umax, Zsum, topidx, Aw);
    gather_kernel<<<CC * KK, 256, 0, stream>>>(feats, topidx, selected);
    fuse_kernel<<<dim3(8, CC), 256, 0, stream>>>(selected, Wv, bv, Aw, fusion);
}